// DyConv_24773371364031
// MI455X (gfx1250) — compile-verified
//
#include <hip/hip_runtime.h>
#include <hip/hip_bf16.h>
#include <math.h>

// Problem constants (from reference): B=48, C=64, H=W=128, K=4, RED=4
#define NB    48
#define NC    64
#define NHW   16384           // 128*128
#define NK    4
#define NHID  16
#define NTILE 128             // spatial columns per block in the GEMM
#define LSTR  140             // LDS row stride (floats): conflict-free for
                              //  - B-frag reads (rows r,r+2 -> +24 banks)
                              //  - output repack (rows r,r+8 -> +32 banks)

typedef float v2f __attribute__((ext_vector_type(2)));
typedef float v8f __attribute__((ext_vector_type(8)));

// Types for the async-to-LDS builtin: per hipcc diagnostic the params are
// pointers to 16B int vectors, global (AS1) src / LDS (AS3) dst.
typedef int v4i __attribute__((__vector_size__(16)));
typedef __attribute__((address_space(1))) v4i as1_v4i;
typedef __attribute__((address_space(3))) v4i as3_v4i;

#if defined(__HIP_DEVICE_COMPILE__) && __has_builtin(__builtin_amdgcn_global_load_async_to_lds_b128)
#define HAVE_ASYNC_LDS 1
#else
#define HAVE_ASYNC_LDS 0
#endif

// ---------------------------------------------------------------------------
// Kernel 1: g[b][c] = mean over H,W of x[b][c][:][:]
// 3072 blocks x 256 threads; 16 float4 loads per thread, fully coalesced.
// ---------------------------------------------------------------------------
__global__ __launch_bounds__(256) void dyconv_mean(const float* __restrict__ x,
                                                   float* __restrict__ g) {
  __shared__ float red[256];
  const int bc  = blockIdx.x;               // 0 .. B*C-1
  const int t   = threadIdx.x;
  const float* p = x + (size_t)bc * NHW;
  float s = 0.0f;
#pragma unroll
  for (int i = 0; i < 16; ++i) {
    const float4 v = *(const float4*)(p + (size_t)(i * 256 + t) * 4);
    s += v.x + v.y + v.z + v.w;
  }
  red[t] = s;
  __syncthreads();
#pragma unroll
  for (int off = 128; off > 0; off >>= 1) {
    if (t < off) red[t] += red[t + off];
    __syncthreads();
  }
  if (t == 0) g[bc] = red[0] * (1.0f / 16384.0f);
}

// ---------------------------------------------------------------------------
// Kernel 2: gating MLP + softmax, then write Wmix pre-swizzled into WMMA
// A-fragment layout:
//   frag[b][ks][mt][lane*2+v] = Wmix[b][mt*16 + lane%16][ks*4 + 2*(lane/16) + v]
// (matches V_WMMA_F32_16X16X4_F32 A-matrix VGPR striping), so the GEMM's
// A loads are coalesced 8B-per-lane loads. 48 blocks x 64 threads.
// ---------------------------------------------------------------------------
__global__ __launch_bounds__(64) void dyconv_gate(const float* __restrict__ g,
                                                  const float* __restrict__ experts,
                                                  const float* __restrict__ fc1,
                                                  const float* __restrict__ fc2,
                                                  float* __restrict__ frag) {
  __shared__ float gs[NC];
  __shared__ float hs[NHID];
  __shared__ float alpha[NK];
  const int b = blockIdx.x;
  const int t = threadIdx.x;

  gs[t] = g[b * NC + t];
  __syncthreads();

  if (t < NHID) {
    float s = 0.0f;
#pragma unroll
    for (int c = 0; c < NC; ++c) s += gs[c] * fc1[t * NC + c];
    hs[t] = s > 0.0f ? s : 0.0f;              // relu
  }
  __syncthreads();

  if (t < NK) {
    float s = 0.0f;
#pragma unroll
    for (int j = 0; j < NHID; ++j) s += hs[j] * fc2[t * NHID + j];
    alpha[t] = s;                              // logits for now
  }
  __syncthreads();

  if (t == 0) {
    const float m  = fmaxf(fmaxf(alpha[0], alpha[1]), fmaxf(alpha[2], alpha[3]));
    const float e0 = expf(alpha[0] - m), e1 = expf(alpha[1] - m);
    const float e2 = expf(alpha[2] - m), e3 = expf(alpha[3] - m);
    const float inv = 1.0f / (e0 + e1 + e2 + e3);
    alpha[0] = e0 * inv; alpha[1] = e1 * inv;
    alpha[2] = e2 * inv; alpha[3] = e3 * inv;
  }
  __syncthreads();

  const float a0 = alpha[0], a1 = alpha[1], a2 = alpha[2], a3 = alpha[3];
  float* fb = frag + (size_t)b * 4096;         // 16 ks * 4 mt * 64
  for (int e = 0; e < 64; ++e) {
    const int flat = e * 64 + t;               // coalesced across threads
    const int v    = flat & 1;
    const int lane = (flat >> 1) & 31;
    const int mt   = (flat >> 6) & 3;
    const int ks   = flat >> 8;
    const int o    = mt * 16 + (lane & 15);
    const int i    = ks * 4 + 2 * (lane >> 4) + v;
    const int oi   = o * NC + i;
    const float w  = a0 * experts[0 * NC * NC + oi] + a1 * experts[1 * NC * NC + oi] +
                     a2 * experts[2 * NC * NC + oi] + a3 * experts[3 * NC * NC + oi];
    fb[flat] = w;
  }
}

// ---------------------------------------------------------------------------
// Kernel 3: y[b] = Wmix[b] @ x[b] via V_WMMA_F32_16X16X4_F32.
// Grid (48, 128); block 256 (8 waves). Wave w owns M=64, N-slice [w*16,w*16+16).
// x tile staged to LDS with async row copies (512B contiguous each).
// ---------------------------------------------------------------------------
__global__ __launch_bounds__(256) void dyconv_gemm(const float* __restrict__ x,
                                                   const float* __restrict__ frag,
                                                   float* __restrict__ y) {
  __shared__ float tile[NC * LSTR];            // 35,840 B

  const int b    = blockIdx.x;
  const int nb   = blockIdx.y * NTILE;
  const int tid  = threadIdx.x;
  const int w    = tid >> 5;                   // wave 0..7
  const int l    = tid & 31;                   // lane 0..31
  const int hl   = l >> 4;                     // half-wave 0/1
  const int c16  = l & 15;

  const float* xb = x + (size_t)b * NC * NHW + nb;

  // ---- stage x[b][0:64][nb:nb+128]: wave w copies rows w*8 .. w*8+7,
  // ---- one full 512B row per async instruction (lane l -> 16B at col l*4).
#if HAVE_ASYNC_LDS
#pragma unroll
  for (int j = 0; j < 8; ++j) {
    const int row = w * 8 + j;
    as1_v4i* src = (as1_v4i*)(size_t)(const void*)(xb + (size_t)row * NHW + l * 4);
    as3_v4i* dst = (as3_v4i*)(unsigned)(size_t)(&tile[row * LSTR + l * 4]);
    __builtin_amdgcn_global_load_async_to_lds_b128(src, dst, 0, 0);
  }
#if __has_builtin(__builtin_amdgcn_s_wait_asynccnt)
  __builtin_amdgcn_s_wait_asynccnt(0);
#else
  asm volatile("s_wait_asynccnt 0" ::: "memory");
#endif
#else
#pragma unroll
  for (int j = 0; j < 8; ++j) {
    const int row = w * 8 + j;
    const float4 v = *(const float4*)(xb + (size_t)row * NHW + l * 4);
    *(float4*)(&tile[row * LSTR + l * 4]) = v;
  }
#endif
  __syncthreads();

  // ---- 16 K-steps of K=4, 4 M-tiles each -> 64 WMMA per wave
  v8f acc[4] = {};
  const float* fb   = frag + (size_t)b * 4096;
  const int    wcol = w * 16 + c16;

  for (int ks = 0; ks < 16; ++ks) {
    const int r0 = ks * 4 + 2 * hl;            // B-matrix K rows for this lane
    v2f bfrag;
    bfrag.x = tile[(r0 + 0) * LSTR + wcol];
    bfrag.y = tile[(r0 + 1) * LSTR + wcol];

    const float* fks = fb + ks * 256 + l * 2;  // A-fragments, coalesced 8B/lane
    const v2f a0 = *(const v2f*)(fks + 0 * 64);
    const v2f a1 = *(const v2f*)(fks + 1 * 64);
    const v2f a2 = *(const v2f*)(fks + 2 * 64);
    const v2f a3 = *(const v2f*)(fks + 3 * 64);

    acc[0] = __builtin_amdgcn_wmma_f32_16x16x4_f32(false, a0, false, bfrag, (short)0, acc[0], false, false);
    acc[1] = __builtin_amdgcn_wmma_f32_16x16x4_f32(false, a1, false, bfrag, (short)0, acc[1], false, false);
    acc[2] = __builtin_amdgcn_wmma_f32_16x16x4_f32(false, a2, false, bfrag, (short)0, acc[2], false, false);
    acc[3] = __builtin_amdgcn_wmma_f32_16x16x4_f32(false, a3, false, bfrag, (short)0, acc[3], false, false);
  }

  // ---- repack through LDS so global stores are coalesced float4
  __syncthreads();                             // all B-frag reads done
#pragma unroll
  for (int mt = 0; mt < 4; ++mt) {
#pragma unroll
    for (int v = 0; v < 8; ++v) {
      const int o = mt * 16 + v + 8 * hl;      // C/D layout: VGPR v, halves M +8
      tile[o * LSTR + wcol] = acc[mt][v];
    }
  }
  __syncthreads();

  float* yb = y + (size_t)b * NC * NHW + nb;
#pragma unroll
  for (int it = 0; it < 8; ++it) {
    const int f4  = it * 256 + tid;            // 0..2047
    const int row = f4 >> 5;
    const int c4  = (f4 & 31) * 4;
    const float4 v = *(const float4*)(&tile[row * LSTR + c4]);
    *(float4*)(yb + (size_t)row * NHW + c4) = v;
  }
}

// ---------------------------------------------------------------------------
extern "C" void kernel_launch(void* const* d_in, const int* in_sizes, int n_in,
                              void* d_out, int out_size, void* d_ws, size_t ws_size,
                              hipStream_t stream) {
  (void)in_sizes; (void)n_in; (void)out_size; (void)ws_size;
  const float* x       = (const float*)d_in[0];   // (48,64,128,128)
  const float* experts = (const float*)d_in[1];   // (4,64,64,1,1)
  const float* fc1     = (const float*)d_in[2];   // (16,64)
  const float* fc2     = (const float*)d_in[3];   // (4,16)
  float*       y       = (float*)d_out;           // (48,64,128,128)

  float* frag = (float*)d_ws;                     // 48*4096 floats = 768 KB
  float* g    = frag + (size_t)NB * 4096;         // 48*64 floats

  dyconv_mean<<<NB * NC, 256, 0, stream>>>(x, g);
  dyconv_gate<<<NB, 64, 0, stream>>>(g, experts, fc1, fc2, frag);
  dyconv_gemm<<<dim3(NB, NHW / NTILE), 256, 0, stream>>>(x, frag, y);
}